// YOLOLoss_63797444214828
// MI455X (gfx1250) — compile-verified
//
#include <hip/hip_runtime.h>

typedef __attribute__((ext_vector_type(2))) float v2f;
typedef __attribute__((ext_vector_type(8))) float v8f;

#define C_CH   85
#define N_POS  22743
#define N_BATCH 64
#define N_CELLS 7581            // 76*76 + 38*38 + 19*19
#define TOTAL_THREADS (N_BATCH * N_CELLS)
#define BLOCK  256
#define NBLK   ((TOTAL_THREADS + BLOCK - 1) / BLOCK)   // 1896
#define NQ     7                // m, xy, wh, cls, obj0, obj1, obj2

// anchors[i]/stride per level, pre-divided (stride = 8,16,32)
__constant__ float c_aw[9] = {1.25f, 2.0f,   4.125f,
                              1.875f, 3.875f, 3.6875f,
                              3.625f, 4.875f, 11.65625f};
__constant__ float c_ah[9] = {1.625f, 3.75f,  2.875f,
                              3.8125f, 2.8125f, 7.4375f,
                              2.8125f, 6.1875f, 10.1875f};

__device__ __forceinline__ float sigmoidf(float x) {
  return 1.0f / (1.0f + expf(-x));
}

// matches jnp.clip(p, 1e-12, 1-1e-12) in fp32 (upper bound rounds to 1.0f)
__device__ __forceinline__ float bcef(float p, float t) {
  p = fminf(fmaxf(p, 1e-12f), 1.0f);
  return -(t * logf(p) + (1.0f - t) * logf(1.0f - p));
}

// Horizontal 32-lane sum via V_WMMA_F32_16X16X4_F32 with B = ones.
// A layout (16x4 f32): vgpr0 lanes0-15 -> A[m][0], lanes16-31 -> A[m][2];
// vgpr1 -> K=1,3 (set to 0). With B=1s, every D element of row m is the
// row sum, so sum_j d[j] gives rows 0-7 (+16-23) on lanes 0-15 and rows
// 8-15 (+24-31) on lanes 16-31; one xor-16 shuffle completes the total.
// Caller must ensure full EXEC (all lanes of the wave active).
__device__ __forceinline__ float wave_reduce_wmma(float v) {
  v2f a = {v, 0.0f};
  v2f b = {1.0f, 1.0f};
  v8f c = {0.0f, 0.0f, 0.0f, 0.0f, 0.0f, 0.0f, 0.0f, 0.0f};
  v8f d = __builtin_amdgcn_wmma_f32_16x16x4_f32(false, a, false, b,
                                                (short)0, c, false, false);
  float t = d[0] + d[1] + d[2] + d[3] + d[4] + d[5] + d[6] + d[7];
  t += __shfl_xor(t, 16, 32);
  return t;
}

__global__ void __launch_bounds__(BLOCK)
yolo_main(const float* __restrict__ x, const float* __restrict__ tg,
          float* __restrict__ bs) {
  const int tid = threadIdx.x;
  const int idx = blockIdx.x * BLOCK + tid;

  float s[NQ];
#pragma unroll
  for (int q = 0; q < NQ; ++q) s[q] = 0.0f;

  if (idx < TOTAL_THREADS) {
    const int b = idx / N_CELLS;
    const int cell = idx - b * N_CELLS;
    int level, cellL, off;
    float gridf;
    if (cell < 5776)      { level = 0; cellL = cell;        off = 0;     gridf = 76.0f; }
    else if (cell < 7220) { level = 1; cellL = cell - 5776; off = 17328; gridf = 38.0f; }
    else                  { level = 2; cellL = cell - 7220; off = 21660; gridf = 19.0f; }
    const int n0 = off + cellL * 3;
    const float* xr = x  + ((size_t)b * N_POS + n0) * C_CH;
    const float* tr = tg + ((size_t)b * N_POS + n0) * C_CH;
    __builtin_prefetch(xr, 0, 1);
    __builtin_prefetch(tr, 0, 1);

    float iou[3], xv4[3], tv4[3];
#pragma unroll
    for (int a = 0; a < 3; ++a) {
      const float* xp = xr + a * C_CH;
      const float* tp = tr + a * C_CH;
      const float x0 = xp[0], x1 = xp[1], x2 = xp[2], x3 = xp[3], x4 = xp[4];
      const float t0 = tp[0], t1 = tp[1], t2 = tp[2], t3 = tp[3], t4 = tp[4];

      if (t4 > 0.0f) {            // object mask m
        s[0] += 1.0f;
        s[1] += bcef(sigmoidf(x0), t0) + bcef(sigmoidf(x1), t1);
        s[2] += bcef(x2, t2) + bcef(x3, t3);
        s[3] += bcef(sigmoidf(x4), t4);
      }
      const float aw = c_aw[level * 3 + a], ah = c_ah[level * 3 + a];
      const float gw = t2 * gridf, gh = t3 * gridf;
      const float inter = fminf(gw, aw) * fminf(gh, ah);
      const float uni   = gw * gh + aw * ah - inter;
      iou[a] = inter / (uni + 1e-12f);
      xv4[a] = x4; tv4[a] = t4;
    }
    // first-max argmax over the 3 anchors (jnp.argmax semantics)
    int best = 0; float ibest = iou[0];
    if (iou[1] > ibest) { best = 1; ibest = iou[1]; }
    if (iou[2] > ibest) { best = 2; }

    float so = 0.0f;
#pragma unroll
    for (int a = 0; a < 3; ++a) {
      const float keep = ((iou[a] <= 0.7f) || (a == best)) ? 1.0f : 0.0f;
      so += bcef(sigmoidf(xv4[a] * keep), tv4[a] * keep);
    }
    s[4 + level] += so;
  }

  // ---- block reduction (all 256 threads active: EXEC all-ones for WMMA) ----
  float wt[NQ];
#pragma unroll
  for (int q = 0; q < NQ; ++q) wt[q] = wave_reduce_wmma(s[q]);

  __shared__ float red[8][NQ];
  const int wave = tid >> 5, lane = tid & 31;
  if (lane == 0) {
#pragma unroll
    for (int q = 0; q < NQ; ++q) red[wave][q] = wt[q];
  }
  __syncthreads();
  if (tid < 8) {
    float v = 0.0f;
    if (tid < NQ) {
#pragma unroll
      for (int w = 0; w < 8; ++w) v += red[w][tid];
    }
    bs[(size_t)blockIdx.x * 8 + tid] = v;  // slot 7 = pad (always written)
  }
}

__global__ void __launch_bounds__(BLOCK)
yolo_finalize(const float* __restrict__ bs, float* __restrict__ out) {
  float acc[NQ];
#pragma unroll
  for (int q = 0; q < NQ; ++q) acc[q] = 0.0f;

  for (int i = threadIdx.x; i < NBLK; i += BLOCK) {
#pragma unroll
    for (int q = 0; q < NQ; ++q) acc[q] += bs[(size_t)i * 8 + q];
  }

  float wt[NQ];
#pragma unroll
  for (int q = 0; q < NQ; ++q) wt[q] = wave_reduce_wmma(acc[q]);

  __shared__ float red[8][NQ];
  const int wave = threadIdx.x >> 5, lane = threadIdx.x & 31;
  if (lane == 0) {
#pragma unroll
    for (int q = 0; q < NQ; ++q) red[wave][q] = wt[q];
  }
  __syncthreads();

  if (threadIdx.x == 0) {
    float tot[NQ];
#pragma unroll
    for (int q = 0; q < NQ; ++q) {
      float v = 0.0f;
#pragma unroll
      for (int w = 0; w < 8; ++w) v += red[w][q];
      tot[q] = v;
    }
    const float n_obj = fmaxf(tot[0], 1.0f);
    const float loss = (tot[1] + tot[2]) / (2.0f * n_obj)
                     + tot[3] / n_obj
                     + tot[4] / (64.0f * 17328.0f)
                     + tot[5] / (64.0f * 4332.0f)
                     + tot[6] / (64.0f * 1083.0f);
    out[0] = loss;
  }
}

extern "C" void kernel_launch(void* const* d_in, const int* in_sizes, int n_in,
                              void* d_out, int out_size, void* d_ws, size_t ws_size,
                              hipStream_t stream) {
  (void)in_sizes; (void)n_in; (void)out_size; (void)ws_size;
  const float* x  = (const float*)d_in[0];
  const float* tg = (const float*)d_in[1];
  float* out = (float*)d_out;
  float* bs  = (float*)d_ws;   // NBLK * 8 floats ≈ 60 KB

  yolo_main<<<NBLK, BLOCK, 0, stream>>>(x, tg, bs);
  yolo_finalize<<<1, BLOCK, 0, stream>>>(bs, out);
}